// PEG_Shift_41480794144826
// MI455X (gfx1250) — compile-verified
//
#include <hip/hip_runtime.h>
#include <stdint.h>

// ---------------------------------------------------------------------------
// PEG_Shift depthwise 3x3 conv, quantized (DeepShift pow2 weights, 16.16 acts)
//   x:      (32, 384, 56, 56) f32
//   weight: (384, 1, 3, 3)    f32  -> sign * 2^round(log2(clamp(|w|,2^-14,1)))
//   bias:   (384,)            f32  -> 16.16 fixed point
//   out:    (32, 384, 56, 56) f32
// Memory-bound (~2.25 FLOP/B, ~308 MB): TDM-staged LDS tiles, double buffered.
// ---------------------------------------------------------------------------

#define DIMCH   384
#define HW      56
#define PLANE   (HW * HW)           // 3136 elements = 12544 B per plane
#define NBATCH  32
#define NPLANES (NBATCH * DIMCH)    // 12288 planes
#define BLOCK   128                 // 4 x wave32
#define GRID    1536                // 8 planes per block, grid-stride
#define RPW     14                  // rows per wave (4 waves * 14 = 56)

#if __has_builtin(__builtin_amdgcn_tensor_load_to_lds)
#define USE_TDM 1
#else
#define USE_TDM 0
#endif

typedef unsigned int u32x4 __attribute__((ext_vector_type(4)));
typedef int          i32x4 __attribute__((ext_vector_type(4)));
typedef int          i32x8 __attribute__((ext_vector_type(8)));

// 16.16 fixed point: clip(floor(x * 2^16) * 2^-16, -2^15, 2^15 - 1)
__device__ __forceinline__ float qfix(float v) {
    v = floorf(v * 65536.0f) * (1.0f / 65536.0f);
    return fminf(fmaxf(v, -32768.0f), 32767.0f);
}

#if USE_TDM
// Build a CDNA5 Tensor DMA descriptor (D#) for one contiguous 56x56 f32 tile
// and issue TENSOR_LOAD_TO_LDS.  CDNA5 ISA (8.3/8.4 bit layout).
__device__ __forceinline__ void tdm_load_plane(const float* gsrc, unsigned lds_off) {
    const uint64_t ga = (uint64_t)(uintptr_t)gsrc;

    // Group 0: [1:0]=count=1, [63:32]=lds_addr, [120:64]=global_addr, [127:126]=type=2
    u32x4 g0;
    g0[0] = 1u;
    g0[1] = lds_off;
    g0[2] = (unsigned)ga;
    g0[3] = ((unsigned)(ga >> 32) & 0x01FFFFFFu) | (2u << 30);

    // Group 1: wg_mask=0, data_size=2 (4B), tensor 56x56, tile 56x56, stride0=56
    union { uint64_t q[4]; i32x8 v; } g1;
    const uint64_t dim0 = HW, dim1 = HW, tile0 = HW, tile1 = HW;
    const uint64_t stride0 = HW, stride1 = PLANE;
    g1.q[0] = (2ull << 16) | ((dim0 & 0xFFFFull) << 48);
    g1.q[1] = (dim0 >> 16) | (dim1 << 16) | (tile0 << 48);
    g1.q[2] = tile1 | ((stride0 & 0xFFFFFFFFull) << 32);
    g1.q[3] = (stride0 >> 32) | ((stride1 & 0xFFFFFFFFFFFFull) << 16);

    i32x4 gz = {0, 0, 0, 0};   // 2D tile: groups 2/3 unused
#if defined(__clang_major__) && (__clang_major__ >= 23)
    i32x8 gz8 = {0, 0, 0, 0, 0, 0, 0, 0};
    __builtin_amdgcn_tensor_load_to_lds(g0, g1.v, gz, gz, gz8, 0);
#else
    __builtin_amdgcn_tensor_load_to_lds(g0, g1.v, gz, gz, 0);
#endif
}
#endif

// Load 4 horizontally consecutive, fixed-point-quantized values of LDS row r
// (columns c0-1 .. c0+2), with zero padding outside the 56x56 plane.
__device__ __forceinline__ void load_row4(const float* src, int r, int c0,
                                          float& v0, float& v1, float& v2, float& v3) {
    if (r < 0 || r >= HW) { v0 = v1 = v2 = v3 = 0.0f; return; }
    const float* row = src + r * HW;
    v0 = (c0 > 0)       ? qfix(row[c0 - 1]) : 0.0f;
    v1 = qfix(row[c0]);
    v2 = qfix(row[c0 + 1]);                 // c0 <= 54 -> always in range
    v3 = (c0 + 2 < HW)  ? qfix(row[c0 + 2]) : 0.0f;
}

__global__ __launch_bounds__(BLOCK)
void peg_shift_dwconv(const float* __restrict__ x,
                      const float* __restrict__ weight,
                      const float* __restrict__ bias,
                      float* __restrict__ out) {
    __shared__ float lds[2][PLANE];     // double buffer, 25 KB of 320 KB WGP LDS

    const int tid  = threadIdx.x;
    const int wave = tid >> 5;
    const int lane = tid & 31;
    const int c0   = lane * 2;          // lanes 0..27 cover columns 0..55
    const int r0   = wave * RPW;

#if USE_TDM
    if (wave == 0)
        tdm_load_plane(x + (size_t)blockIdx.x * PLANE, (unsigned)(uintptr_t)&lds[0][0]);
#endif

    int it = 0;
    for (int p = blockIdx.x; p < NPLANES; p += GRID, ++it) {
        const int buf = it & 1;
        const int pn  = p + GRID;

#if USE_TDM
        if (wave == 0) {
            if (pn < NPLANES) {
                tdm_load_plane(x + (size_t)pn * PLANE,
                               (unsigned)(uintptr_t)&lds[buf ^ 1][0]);
                __builtin_amdgcn_s_wait_tensorcnt(1);   // current plane landed
            } else {
                __builtin_amdgcn_s_wait_tensorcnt(0);
            }
        }
#else
        for (int i = tid; i < PLANE; i += BLOCK)
            lds[buf][i] = x[(size_t)p * PLANE + i];
#endif
        __syncthreads();                 // publish lds[buf] to all 4 waves

        // Per-channel pow2-quantized weights + fixed-point bias
        const int ch = p % DIMCH;
        const float* wp = weight + ch * 9;
        float wq[9];
#pragma unroll
        for (int k = 0; k < 9; ++k) {
            const float w  = wp[k];
            const float s  = (w > 0.0f) ? 1.0f : ((w < 0.0f) ? -1.0f : 0.0f);
            float aw = fminf(fmaxf(fabsf(w), 0x1p-14f), 1.0f);
            wq[k] = s * exp2f(rintf(log2f(aw)));
        }
        const float bq = qfix(bias[ch]);

        if (c0 < HW) {
            const float* __restrict__ src = &lds[buf][0];
            float* __restrict__ dst = out + (size_t)p * PLANE;

            float t00, t01, t02, t03;    // row r-1
            float t10, t11, t12, t13;    // row r
            load_row4(src, r0 - 1, c0, t00, t01, t02, t03);
            load_row4(src, r0,     c0, t10, t11, t12, t13);

#pragma unroll 2
            for (int r = r0; r < r0 + RPW; ++r) {
                float t20, t21, t22, t23;              // row r+1
                load_row4(src, r + 1, c0, t20, t21, t22, t23);

                const float o0 = wq[0]*t00 + wq[1]*t01 + wq[2]*t02
                               + wq[3]*t10 + wq[4]*t11 + wq[5]*t12
                               + wq[6]*t20 + wq[7]*t21 + wq[8]*t22 + bq;
                const float o1 = wq[0]*t01 + wq[1]*t02 + wq[2]*t03
                               + wq[3]*t11 + wq[4]*t12 + wq[5]*t13
                               + wq[6]*t21 + wq[7]*t22 + wq[8]*t23 + bq;

                __builtin_nontemporal_store(o0, dst + r * HW + c0);      // streaming
                __builtin_nontemporal_store(o1, dst + r * HW + c0 + 1);

                t00 = t10; t01 = t11; t02 = t12; t03 = t13;
                t10 = t20; t11 = t21; t12 = t22; t13 = t23;
            }
        }
        __syncthreads();                 // lds[buf] free before it's refilled
    }
}

extern "C" void kernel_launch(void* const* d_in, const int* in_sizes, int n_in,
                              void* d_out, int out_size, void* d_ws, size_t ws_size,
                              hipStream_t stream) {
    const float* x = (const float*)d_in[0];
    const float* w = (const float*)d_in[1];
    const float* b = (const float*)d_in[2];
    float* out     = (float*)d_out;
    peg_shift_dwconv<<<dim3(GRID), dim3(BLOCK), 0, stream>>>(x, w, b, out);
}